// ASDHead_71846212927768
// MI455X (gfx1250) — compile-verified
//
#include <hip/hip_runtime.h>

typedef __attribute__((ext_vector_type(2))) float v2f;
typedef __attribute__((ext_vector_type(8))) float v8f;

#define D_MODEL  256
#define D_HIDDEN 128
#define B_SZ     8
#define T_SZ     1024
#define N_SLOTS  64

#define FS_STRIDE 133   // gcd(133,64)=1 -> conflict-free LDS banking
#define SS_STRIDE 133

// ---------------------------------------------------------------------------
// Kernel 1: s_proj[b*64+n, h] = slots[b,n,:] @ W_proj[256:512, h] + b_proj[h]
// One wave32 per 16x16 output tile; K=256 via 64 x V_WMMA_F32_16X16X4_F32.
// ---------------------------------------------------------------------------
__global__ __launch_bounds__(128) void sproj_wmma_kernel(
    const float* __restrict__ slots, const float* __restrict__ W_proj,
    const float* __restrict__ b_proj, float* __restrict__ s_proj)
{
  const int wave = threadIdx.x >> 5;
  const int lane = threadIdx.x & 31;
  const int half = lane >> 4;
  const int l16  = lane & 15;

  const int tile   = blockIdx.x * 4 + wave;  // 0..255
  const int m0     = (tile >> 3) * 16;       // row tile over 512 slot-rows
  const int n0     = (tile & 7) * 16;        // col tile over 128 hidden

  // A: slots flattened (512 x 256), lane holds row m0+l16, K pair at 2*half
  const float* Arow = slots + (size_t)(m0 + l16) * D_MODEL;
  // B: W_s = W_proj rows [256,512), column n0+l16, row stride 128
  const float* Bcol = W_proj + (size_t)D_MODEL * D_HIDDEN + n0 + l16;

  v8f acc = {};
#pragma clang loop unroll_count(8)
  for (int kk = 0; kk < 64; ++kk) {
    const int k = kk * 4 + 2 * half;
    v2f a, bb;
    a.x  = Arow[k];
    a.y  = Arow[k + 1];
    bb.x = Bcol[(size_t)k * D_HIDDEN];
    bb.y = Bcol[(size_t)(k + 1) * D_HIDDEN];
    acc = __builtin_amdgcn_wmma_f32_16x16x4_f32(false, a, false, bb,
                                                (short)0, acc, false, false);
  }

  const float bias = b_proj[n0 + l16];
  float* outc = s_proj + (size_t)n0 + l16;
#pragma unroll
  for (int v = 0; v < 8; ++v) {
    const int row = m0 + v + 8 * half;
    outc[(size_t)row * D_HIDDEN] = acc[v] + bias;
  }
}

// ---------------------------------------------------------------------------
// Kernel 2 (fused): per block = (b, 16 t-rows).
//   Phase A: stage s_proj[b] (64x128) + w_head into LDS.
//   Phase B: each of 8 waves computes one 16x16 h-slice of f_proj with fp32
//            WMMA (K=256) straight from features/W_proj, deposits into LDS.
//   Phase C: 1024 (t,n) outputs; relu(f+s) dot w_head on the VALU.
// ---------------------------------------------------------------------------
__global__ __launch_bounds__(256) void fused_head_kernel(
    const float* __restrict__ features, const float* __restrict__ W_proj,
    const float* __restrict__ s_proj, const float* __restrict__ w_head,
    const float* __restrict__ b_head_p, float* __restrict__ out)
{
  __shared__ float s_s[N_SLOTS * SS_STRIDE];   // ~34 KB
  __shared__ float f_s[16 * FS_STRIDE];        // ~8.5 KB
  __shared__ float w_s[D_HIDDEN];

  const int tid    = threadIdx.x;
  const int b      = blockIdx.x >> 6;   // 0..7
  const int t0     = (blockIdx.x & 63) * 16;

  // Phase A: stage s_proj for this batch
  for (int i = tid; i < N_SLOTS * D_HIDDEN; i += 256) {
    const int r = i >> 7, c = i & 127;
    s_s[r * SS_STRIDE + c] = s_proj[((size_t)b * N_SLOTS + r) * D_HIDDEN + c];
  }
  if (tid < D_HIDDEN) w_s[tid] = w_head[tid];

  // Phase B: f_proj tile via WMMA; wave owns h-cols [16*wave, 16*wave+16)
  const int wave = tid >> 5;
  const int lane = tid & 31;
  const int half = lane >> 4;
  const int l16  = lane & 15;
  const int h0   = wave * 16;

  const float* Arow = features + ((size_t)b * T_SZ + t0 + l16) * D_MODEL;
  const float* Bcol = W_proj + h0 + l16;   // W_f rows [0,256)

  v8f acc = {};
#pragma clang loop unroll_count(8)
  for (int kk = 0; kk < 64; ++kk) {
    const int k = kk * 4 + 2 * half;
    v2f a, bb;
    a.x  = Arow[k];
    a.y  = Arow[k + 1];
    bb.x = Bcol[(size_t)k * D_HIDDEN];
    bb.y = Bcol[(size_t)(k + 1) * D_HIDDEN];
    acc = __builtin_amdgcn_wmma_f32_16x16x4_f32(false, a, false, bb,
                                                (short)0, acc, false, false);
  }
#pragma unroll
  for (int v = 0; v < 8; ++v)
    f_s[(v + 8 * half) * FS_STRIDE + h0 + l16] = acc[v];

  __syncthreads();

  // Phase C: out[b, t0+t, n] = sum_h relu(f+s)*w + b_head
  const float b_head = *b_head_p;
#pragma unroll
  for (int p = 0; p < 4; ++p) {
    const int idx = p * 256 + tid;      // 0..1023
    const int t   = idx >> 6;           // 0..15 (uniform per wave -> f broadcast)
    const int n   = idx & 63;           // lane-varying -> conflict-free s reads
    const float* frow = &f_s[t * FS_STRIDE];
    const float* srow = &s_s[n * SS_STRIDE];
    float a0 = 0.f, a1 = 0.f;
#pragma unroll 8
    for (int h = 0; h < D_HIDDEN; h += 2) {
      a0 += fmaxf(frow[h]     + srow[h],     0.f) * w_s[h];
      a1 += fmaxf(frow[h + 1] + srow[h + 1], 0.f) * w_s[h + 1];
    }
    out[(((size_t)b * T_SZ) + t0 + t) * N_SLOTS + n] = a0 + a1 + b_head;
  }
}

extern "C" void kernel_launch(void* const* d_in, const int* in_sizes, int n_in,
                              void* d_out, int out_size, void* d_ws, size_t ws_size,
                              hipStream_t stream) {
  (void)in_sizes; (void)n_in; (void)out_size; (void)ws_size;
  const float* features = (const float*)d_in[0];
  const float* slots    = (const float*)d_in[1];
  const float* W_proj   = (const float*)d_in[2];
  const float* b_proj   = (const float*)d_in[3];
  const float* w_head   = (const float*)d_in[4];
  const float* b_head   = (const float*)d_in[5];
  float* out    = (float*)d_out;
  float* s_proj = (float*)d_ws;   // 512*128*4 = 256 KB scratch

  // 256 tiles of s_proj, 4 waves per 128-thread block
  sproj_wmma_kernel<<<64, 128, 0, stream>>>(slots, W_proj, b_proj, s_proj);
  // 8 batches x 64 t-tiles, 256 threads (8 wave32) each
  fused_head_kernel<<<B_SZ * (T_SZ / 16), 256, 0, stream>>>(
      features, W_proj, s_proj, w_head, b_head, out);
}